// AutoCorrelationAttention_41051297415916
// MI455X (gfx1250) — compile-verified
//
#include <hip/hip_runtime.h>
#include <math.h>

// ---------------------------------------------------------------------------
// AutoCorrelation attention (Autoformer) for MI455X / gfx1250.
//   B=16, L=4096, d=512, fp32.
//   Stage 1: per-channel circular cross-correlation via radix-64 GEMM-FFT
//            (fused complex 64x64x64 fp32 GEMMs on V_WMMA_F32_16X16X4_F32,
//             DFT-64 matrix shared by all 8192 channels)
//   Stage 2: top-16 + softmax per channel
//   Stage 3: weighted gather of 16 rolled V copies
//
// LDS planes use a K-pair interleaved layout so every WMMA fragment pair is
// contiguous:  midx(r,c) = (r>>1)*PROW + 2*c + (r&1)
//   B fragment (rows k0,k0+1; k0 even) -> adjacent floats -> one ds_load_b64
//   A fragment (cols k0,k0+1)          -> 8B apart       -> one ds_load_2addr
// ---------------------------------------------------------------------------

typedef __attribute__((ext_vector_type(2))) float v2f;
typedef __attribute__((ext_vector_type(8))) float v8f;

#define LSEQ   4096
#define DMODEL 512
#define NBATCH 16
#define TOPK   16
#define PI_F   3.14159265358979323846f

#define ROWS   68                 // logical row stride (floats), padded vs 64
#define PROW   (2 * ROWS)         // interleaved row-pair stride = 136 floats
#define PLANE  (64 * ROWS)        // 4352 floats per 64x64 matrix plane
#define SMEM_FLOATS (9 * PLANE)   // 39168 floats = 153 KB dynamic LDS (2 blocks / 320KB WGP)

__device__ __forceinline__ int midx(int r, int c)
{
    return ((r >> 1) * PROW) + (c << 1) + (r & 1);
}

// fragment loaders (CDNA5 ISA 7.12.2 layouts):
//   A (16x4 f32):  lanes 0-15 row=lane, VGPR{0,1}=K{0,1}; lanes 16-31 K{2,3}
//   B (4x16 f32):  VGPR v holds row v+2*half, col = lane&15
//   C/D (16x16):   VGPR v holds row v+8*half, col = lane&15
#define LOAD_A(dst, off) do { (dst).x = smem[(off) + abase + (k0 << 1)];      \
                              (dst).y = smem[(off) + abase + (k0 << 1) + 2]; } while (0)
#define LOAD_B(dst, off, cofs) do { const int _b = (off) + kp + bcol + ((cofs) << 1); \
                                    (dst).x = smem[_b]; (dst).y = smem[_b + 1]; } while (0)
#define WMMA(acc, a, b) \
    (acc) = __builtin_amdgcn_wmma_f32_16x16x4_f32(false, (a), false, (b), (short)0, (acc), false, false)

// ---------------------------------------------------------------------------
// Fused complex 64x64x64 GEMM, A-side carries the negated-imag plane:
//   Cr = Ar@Br + An@Bi        (An = -Ai, precomputed in LDS)
//   Ci = Ar@Bi + Ai@Br        (only if CIMAG)
// AIMAG=false degenerates to real A:  Cr = Ar@Br ; Ci = Ar@Bi.
// 4 waves; wave w owns C rows [16w,16w+16); 4 n-tiles resident; overwrite C.
// ---------------------------------------------------------------------------
template <bool AIMAG, bool CIMAG>
__device__ __forceinline__ void cgemmA(int oAr, int oAi, int oAn,
                                       int oBr, int oBi, int oCr, int oCi)
{
    extern __shared__ float smem[];
    const int lane  = threadIdx.x & 31;
    const int wv    = threadIdx.x >> 5;
    const int half  = lane >> 4;
    const int ln    = lane & 15;
    const int abase = (8 * wv + (ln >> 1)) * PROW + (ln & 1);  // row 16*wv+ln
    const int bcol  = ln << 1;

    v8f z = {0.f,0.f,0.f,0.f,0.f,0.f,0.f,0.f};
    v8f cr0 = z, cr1 = z, cr2 = z, cr3 = z;
    v8f ci0 = z, ci1 = z, ci2 = z, ci3 = z;

#pragma unroll
    for (int kk = 0; kk < 16; ++kk) {
        const int k0 = 4 * kk + 2 * half;
        const int kp = (k0 >> 1) * PROW;
        v2f aR, aI, aN;
        LOAD_A(aR, oAr);
        if (AIMAG)          LOAD_A(aN, oAn);
        if (AIMAG && CIMAG) LOAD_A(aI, oAi);

#define CGA_TILE(CR, CI, COFS)                                     \
        {                                                          \
            v2f bR, bI;                                            \
            LOAD_B(bR, oBr, COFS);                                 \
            WMMA(CR, aR, bR);                                      \
            if (AIMAG || CIMAG) LOAD_B(bI, oBi, COFS);             \
            if (AIMAG) WMMA(CR, aN, bI);                           \
            if (CIMAG) { WMMA(CI, aR, bI);                         \
                         if (AIMAG) WMMA(CI, aI, bR); }            \
        }
        CGA_TILE(cr0, ci0,  0)
        CGA_TILE(cr1, ci1, 16)
        CGA_TILE(cr2, ci2, 32)
        CGA_TILE(cr3, ci3, 48)
#undef CGA_TILE
    }

#pragma unroll
    for (int v = 0; v < 8; ++v) {
        // row m = 16*wv + v + 8*half, col = ln (+0/16/32/48)
        const int r = (8 * wv + 4 * half + (v >> 1)) * PROW + (v & 1) + (ln << 1);
        smem[oCr + r +  0] = cr0[v]; smem[oCr + r + 32] = cr1[v];
        smem[oCr + r + 64] = cr2[v]; smem[oCr + r + 96] = cr3[v];
        if (CIMAG) {
            smem[oCi + r +  0] = ci0[v]; smem[oCi + r + 32] = ci1[v];
            smem[oCi + r + 64] = ci2[v]; smem[oCi + r + 96] = ci3[v];
        }
    }
}

// ---------------------------------------------------------------------------
// Fused complex 64x64x64 GEMM, B-side carries the negated-imag plane:
//   Cr = Ar@Br + Ai@Bn        (Bn = -Bi, precomputed in LDS)
//   Ci = Ar@Bi + Ai@Br
// ---------------------------------------------------------------------------
__device__ __forceinline__ void cgemmB(int oAr, int oAi,
                                       int oBr, int oBi, int oBn,
                                       int oCr, int oCi)
{
    extern __shared__ float smem[];
    const int lane  = threadIdx.x & 31;
    const int wv    = threadIdx.x >> 5;
    const int half  = lane >> 4;
    const int ln    = lane & 15;
    const int abase = (8 * wv + (ln >> 1)) * PROW + (ln & 1);
    const int bcol  = ln << 1;

    v8f z = {0.f,0.f,0.f,0.f,0.f,0.f,0.f,0.f};
    v8f cr0 = z, cr1 = z, cr2 = z, cr3 = z;
    v8f ci0 = z, ci1 = z, ci2 = z, ci3 = z;

#pragma unroll
    for (int kk = 0; kk < 16; ++kk) {
        const int k0 = 4 * kk + 2 * half;
        const int kp = (k0 >> 1) * PROW;
        v2f aR, aI;
        LOAD_A(aR, oAr);
        LOAD_A(aI, oAi);

#define CGB_TILE(CR, CI, COFS)                                     \
        {                                                          \
            v2f bR, bI, bN;                                        \
            LOAD_B(bR, oBr, COFS);                                 \
            LOAD_B(bI, oBi, COFS);                                 \
            LOAD_B(bN, oBn, COFS);                                 \
            WMMA(CR, aR, bR); WMMA(CR, aI, bN);                    \
            WMMA(CI, aR, bI); WMMA(CI, aI, bR);                    \
        }
        CGB_TILE(cr0, ci0,  0)
        CGB_TILE(cr1, ci1, 16)
        CGB_TILE(cr2, ci2, 32)
        CGB_TILE(cr3, ci3, 48)
#undef CGB_TILE
    }

#pragma unroll
    for (int v = 0; v < 8; ++v) {
        const int r = (8 * wv + 4 * half + (v >> 1)) * PROW + (v & 1) + (ln << 1);
        smem[oCr + r +  0] = cr0[v]; smem[oCr + r + 32] = cr1[v];
        smem[oCr + r + 64] = cr2[v]; smem[oCr + r + 96] = cr3[v];
        smem[oCi + r +  0] = ci0[v]; smem[oCi + r + 32] = ci1[v];
        smem[oCi + r + 64] = ci2[v]; smem[oCi + r + 96] = ci3[v];
    }
}

// ---------------------------------------------------------------------------
// Four-step length-4096 DFT (N = 64*64), all planes in LDS.
//   input:  X[n1][n2] = x[n1 + 64*n2]
//   stage1: Y = X @ F64 ; twiddle Y[n1,k2] *= w4096^{n1 k2} ; stage2: Z = F64 @ Y
//   Z[k1][k2] = spectrum at natural index 64*k1+k2 ; result left in X planes.
// If !IMAG_OUT only the real plane of Z is produced.
// ---------------------------------------------------------------------------
template <bool COMPLEX_IN, bool IMAG_OUT>
__device__ __forceinline__ void fwd4096(int oXr, int oXi, int oYr, int oYi,
                                        int oFr, int oFi, int oFin)
{
    extern __shared__ float smem[];

    // stage 1: Y = X @ F
    if (COMPLEX_IN) cgemmB(oXr, oXi, oFr, oFi, oFin, oYr, oYi);
    else            cgemmA<false, true>(oXr, 0, 0, oFr, oFi, oYr, oYi);
    __syncthreads();

    // twiddle: Y[n1,k2] *= (cos - i sin)(2*pi*n1*k2/4096)
    for (int i = threadIdx.x; i < 4096; i += 128) {
        const int n1 = i >> 6, k2 = i & 63;
        const int t = midx(n1, k2);
        const float th = (2.0f * PI_F / 4096.0f) * (float)((n1 * k2) & 4095);
        float s, c;
        __sincosf(th, &s, &c);
        const float yr = smem[oYr + t], yi = smem[oYi + t];
        smem[oYr + t] = yr * c + yi * s;
        smem[oYi + t] = yi * c - yr * s;
    }
    __syncthreads();

    // stage 2: Z = F @ Y  (into X planes); F is the (complex) A operand.
    cgemmA<true, IMAG_OUT>(oFr, oFi, oFin, oYr, oYi, oXr, oXi);
}

// ---------------------------------------------------------------------------
// Stage 1 kernel: one block (128 threads, 4 waves) per (b,d) channel.
// Rxx[c, :] = Re(ifft(fft(q) * conj(fft(k))))
// ---------------------------------------------------------------------------
__global__ void __launch_bounds__(128)
autocorr_fft_kernel(const float* __restrict__ Q,
                    const float* __restrict__ K,
                    float* __restrict__ R)
{
    extern __shared__ float smem[];
    const int oFr  = 0 * PLANE, oFi = 1 * PLANE, oFin = 2 * PLANE;
    const int oXr  = 3 * PLANE, oXi = 4 * PLANE;
    const int oYr  = 5 * PLANE, oYi = 6 * PLANE;
    const int oQr  = 7 * PLANE, oQi = 8 * PLANE;

    const int c   = blockIdx.x;          // channel index, d fastest (L2 line sharing)
    const int d   = c & (DMODEL - 1);
    const int b   = c >> 9;              // c / 512
    const int tid = threadIdx.x;

    // DFT-64 matrix: F[j][k] = w64^{jk} = cos - i sin ; Fin = -Fi = +sin
    for (int i = tid; i < 4096; i += 128) {
        const int j = i >> 6, k = i & 63;
        const int t = midx(j, k);
        const float th = (2.0f * PI_F / 64.0f) * (float)((j * k) & 63);
        float s, cth;
        __sincosf(th, &s, &cth);
        smem[oFr  + t] = cth;
        smem[oFi  + t] = -s;
        smem[oFin + t] = s;
    }
    __syncthreads();

    // ---- fft(q) -> X planes, saved to Q planes
    for (int n = tid; n < LSEQ; n += 128) {
        const float v = Q[(size_t)(b * LSEQ + n) * DMODEL + d];
        smem[oXr + midx(n & 63, n >> 6)] = v;   // X[n1][n2], n = n1 + 64*n2
    }
    __syncthreads();
    fwd4096<false, true>(oXr, oXi, oYr, oYi, oFr, oFi, oFin);
    __syncthreads();
    for (int i = tid; i < 4096; i += 128) {
        const int t = midx(i >> 6, i & 63);
        smem[oQr + t] = smem[oXr + t];
        smem[oQi + t] = smem[oXi + t];
    }
    __syncthreads();

    // ---- fft(k) -> X planes
    for (int n = tid; n < LSEQ; n += 128) {
        const float v = K[(size_t)(b * LSEQ + n) * DMODEL + d];
        smem[oXr + midx(n & 63, n >> 6)] = v;
    }
    __syncthreads();
    fwd4096<false, true>(oXr, oXi, oYr, oYi, oFr, oFi, oFin);
    __syncthreads();

    // ---- P = Qhat * conj(Khat); stage conj(P) (transposed) into Y planes.
    // ifft(P) = conj(fft(conj(P))) / N ; we only need the real part.
    for (int m = tid; m < 4096; m += 128) {
        const int t = midx(m >> 6, m & 63);         // natural-order index m
        const float qr = smem[oQr + t], qi = smem[oQi + t];
        const float kr = smem[oXr + t], ki = smem[oXi + t];
        const float pr = qr * kr + qi * ki;
        const float pi = qi * kr - qr * ki;
        const int u = midx(m & 63, m >> 6);         // X-layout for next FFT
        smem[oYr + u] = pr;
        smem[oYi + u] = -pi;
    }
    __syncthreads();
    // forward transform of conj(P): input Y planes, scratch X planes,
    // real output lands back in the Y-real plane.
    fwd4096<true, false>(oYr, oYi, oXr, oXi, oFr, oFi, oFin);
    __syncthreads();

    const float inv = 1.0f / 4096.0f;
    for (int m = tid; m < 4096; m += 128) {
        R[(size_t)c * LSEQ + m] = smem[oYr + midx(m >> 6, m & 63)] * inv;
    }
}

// ---------------------------------------------------------------------------
// Stage 2: top-16 + softmax per channel. One 256-thread block per channel.
// ---------------------------------------------------------------------------
__global__ void __launch_bounds__(256)
topk_softmax_kernel(const float* __restrict__ R,
                    float* __restrict__ Wk,
                    int* __restrict__ Ik)
{
    __shared__ float sv[LSEQ];
    __shared__ float rmax[256];
    __shared__ int   ridx[256];
    __shared__ float wv[TOPK];
    __shared__ int   wi[TOPK];

    const int c = blockIdx.x, tid = threadIdx.x;
    for (int i = tid; i < LSEQ; i += 256) sv[i] = R[(size_t)c * LSEQ + i];
    __syncthreads();

    for (int j = 0; j < TOPK; ++j) {
        float m = -INFINITY;
        int mi = 0;
#pragma unroll
        for (int t = 0; t < 16; ++t) {
            const int idx = tid * 16 + t;
            const float v = sv[idx];
            if (v > m) { m = v; mi = idx; }
        }
        rmax[tid] = m; ridx[tid] = mi;
        __syncthreads();
        for (int s = 128; s > 0; s >>= 1) {
            if (tid < s) {
                const float a = rmax[tid], bb = rmax[tid + s];
                if (bb > a || (bb == a && ridx[tid + s] < ridx[tid])) {
                    rmax[tid] = bb; ridx[tid] = ridx[tid + s];
                }
            }
            __syncthreads();
        }
        if (tid == 0) {
            wv[j] = rmax[0];
            wi[j] = ridx[0];
            sv[ridx[0]] = -INFINITY;   // mask winner
        }
        __syncthreads();
    }

    if (tid == 0) {
        const float mx = wv[0];        // first extracted == max
        float e[TOPK], s = 0.f;
        for (int j = 0; j < TOPK; ++j) { e[j] = expf(wv[j] - mx); s += e[j]; }
        const float invs = 1.f / s;
        for (int j = 0; j < TOPK; ++j) {
            Wk[(size_t)c * TOPK + j] = e[j] * invs;
            Ik[(size_t)c * TOPK + j] = wi[j];
        }
    }
}

// ---------------------------------------------------------------------------
// Stage 3: A[b,l,d] = sum_j Wk[b,d,j] * V[b,(l+Ik[b,d,j])%L,d]
// grid: (B * L/16, 2), block 256 over d; each thread handles 16 l values.
// V per batch is 8 MB -> L2-resident; gathers hit L2.
// ---------------------------------------------------------------------------
__global__ void __launch_bounds__(256)
gather_combine_kernel(const float* __restrict__ V,
                      const float* __restrict__ Wk,
                      const int* __restrict__ Ik,
                      float* __restrict__ Out)
{
    const int tid = threadIdx.x;
    const int d   = blockIdx.y * 256 + tid;
    const int b   = blockIdx.x >> 8;            // 256 l-tiles per batch
    const int l0  = (blockIdx.x & 255) * 16;
    const int c   = b * DMODEL + d;

    float w[TOPK];
    int   off[TOPK];
#pragma unroll
    for (int j = 0; j < TOPK; ++j) {
        w[j]   = Wk[(size_t)c * TOPK + j];
        off[j] = Ik[(size_t)c * TOPK + j];
    }

    const size_t base = (size_t)b * LSEQ;
    for (int t = 0; t < 16; ++t) {
        const int l = l0 + t;
        float acc = 0.f;
#pragma unroll
        for (int j = 0; j < TOPK; ++j) {
            const int src = (l + off[j]) & (LSEQ - 1);
            acc += w[j] * V[(base + src) * DMODEL + d];
        }
        Out[(base + l) * DMODEL + d] = acc;
    }
}

// ---------------------------------------------------------------------------
extern "C" void kernel_launch(void* const* d_in, const int* in_sizes, int n_in,
                              void* d_out, int out_size, void* d_ws, size_t ws_size,
                              hipStream_t stream)
{
    const float* Q = (const float*)d_in[0];
    const float* K = (const float*)d_in[1];
    const float* V = (const float*)d_in[2];
    float* Out = (float*)d_out;

    // workspace: Rxx [8192 x 4096] f32, Wk [8192 x 16] f32, Ik [8192 x 16] i32
    float* R = (float*)d_ws;
    const size_t relems = (size_t)NBATCH * DMODEL * LSEQ;
    float* Wk = R + relems;
    int*   Ik = (int*)(Wk + (size_t)NBATCH * DMODEL * TOPK);

    const int nchan = NBATCH * DMODEL;   // 8192

    autocorr_fft_kernel<<<nchan, 128, SMEM_FLOATS * sizeof(float), stream>>>(Q, K, R);
    topk_softmax_kernel<<<nchan, 256, 0, stream>>>(R, Wk, Ik);

    dim3 grid(NBATCH * (LSEQ / 16), 2);
    gather_combine_kernel<<<grid, 256, 0, stream>>>(V, Wk, Ik, Out);
}